// HRDPRec_50165218017399
// MI455X (gfx1250) — compile-verified
//
#include <hip/hip_runtime.h>
#include <hip/hip_bf16.h>
#include <math.h>

typedef float v2f __attribute__((ext_vector_type(2)));
typedef float v8f __attribute__((ext_vector_type(8)));

#define KCHUNK 384   // 384*32*4 = 48 KB LDS per block

#define WMMA_F32X4(A_, B_, C_) \
  __builtin_amdgcn_wmma_f32_16x16x4_f32(false, (A_), false, (B_), (short)0, (C_), false, false)

// ---------------- elementwise helpers ----------------
__global__ __launch_bounds__(256) void hrd_zero(float* p, int n) {
  int i = blockIdx.x * 256 + threadIdx.x;
  if (i < n) p[i] = 0.0f;
}
__global__ __launch_bounds__(256) void hrd_copy(float* dst, const float* src, int n) {
  int i = blockIdx.x * 256 + threadIdx.x;
  if (i < n) dst[i] = src[i];
}
__global__ __launch_bounds__(256) void hrd_mean4(float* out, const float* a, const float* b,
                                                 const float* c, const float* d, int n) {
  int i = blockIdx.x * 256 + threadIdx.x;
  if (i < n) out[i] = 0.25f * (a[i] + b[i] + c[i] + d[i]);
}
__global__ __launch_bounds__(256) void hrd_sum3(float* out, const float* a, const float* b,
                                                const float* c, int n) {
  int i = blockIdx.x * 256 + threadIdx.x;
  if (i < n) out[i] = a[i] + b[i] + c[i];
}

// ---------------- COO SpMM: out[r,:] += v * x[c,:], D=32, wave per nnz ----------------
__global__ __launch_bounds__(256) void hrd_spmm(const int* __restrict__ rows,
                                                const int* __restrict__ cols,
                                                const float* __restrict__ vals,
                                                const float* __restrict__ x,
                                                float* __restrict__ out, int nnz) {
  int lane = threadIdx.x & 31;
  int wid  = (blockIdx.x * blockDim.x + threadIdx.x) >> 5;
  int nw   = (gridDim.x * blockDim.x) >> 5;
  for (int e = wid; e < nnz; e += nw) {
    int r = rows[e];
    int c = cols[e];
    float v = vals[e];
    float xv = x[(size_t)c * 32 + lane];
    atomicAdd(&out[(size_t)r * 32 + lane], v * xv);
  }
}

// ---------------- dense C[M,32] = A[M,K] @ X[K,32] via V_WMMA_F32_16X16X4_F32 ----------
// Wave tiles: 16(M) x 16(N); block = 8 waves = 4 M-tiles x 2 N-tiles = 64 x 32 output.
// X chunk staged in LDS; A streamed from HBM as float2 per lane (256B fresh A per wmma).
// K-loop unrolled x4 so 4 A loads are in flight per wave before the WMMA chain.
__global__ __launch_bounds__(256) void hrd_mm_wmma(const float* __restrict__ A,
                                                   const float* __restrict__ X,
                                                   float* __restrict__ C, int M, int K) {
  __shared__ float ldsX[KCHUNK * 32];
  const int tid  = threadIdx.x;
  const int lane = tid & 31;
  const int wave = tid >> 5;                 // 0..7
  const int mt   = blockIdx.x * 4 + (wave >> 1);
  const int m0   = mt * 16;
  const int n0   = (wave & 1) * 16;
  const int half = lane >> 4;                // 0 / 1
  const int l15  = lane & 15;
  const bool active = (m0 < M);              // wave-uniform

  v8f acc = {};
  for (int kc = 0; kc < K; kc += KCHUNK) {
    int klen = (K - kc < KCHUNK) ? (K - kc) : KCHUNK;   // multiple of 4
    // cooperative copy of contiguous X[kc:kc+klen, :] into LDS (float4)
    const float4* src = (const float4*)(X + (size_t)kc * 32);
    float4* dst = (float4*)ldsX;
    int n4 = klen * 8;
    for (int i = tid; i < n4; i += 256) dst[i] = src[i];
    __syncthreads();
    if (active) {
      const float* Arow = A + (size_t)(m0 + l15) * K + kc + 2 * half;
      // prefetch the next A chunk for this row while we compute on this one
      if (kc + KCHUNK < K) __builtin_prefetch(Arow + KCHUNK, 0, 0);
      const float* Bp = ldsX + (2 * half) * 32 + n0 + l15;   // b.x = Bp[kk*32], b.y = Bp[kk*32+32]
      int kk = 0;
      int klen16 = klen & ~15;
      for (; kk < klen16; kk += 16) {
        v2f a0 = *(const v2f*)(Arow + kk);
        v2f a1 = *(const v2f*)(Arow + kk + 4);
        v2f a2 = *(const v2f*)(Arow + kk + 8);
        v2f a3 = *(const v2f*)(Arow + kk + 12);
        v2f b0, b1, b2, b3;
        b0.x = Bp[(kk +  0) * 32]; b0.y = Bp[(kk +  0) * 32 + 32];
        b1.x = Bp[(kk +  4) * 32]; b1.y = Bp[(kk +  4) * 32 + 32];
        b2.x = Bp[(kk +  8) * 32]; b2.y = Bp[(kk +  8) * 32 + 32];
        b3.x = Bp[(kk + 12) * 32]; b3.y = Bp[(kk + 12) * 32 + 32];
        acc = WMMA_F32X4(a0, b0, acc);
        acc = WMMA_F32X4(a1, b1, acc);
        acc = WMMA_F32X4(a2, b2, acc);
        acc = WMMA_F32X4(a3, b3, acc);
      }
      for (; kk < klen; kk += 4) {
        v2f a = *(const v2f*)(Arow + kk);
        v2f b;
        b.x = Bp[kk * 32];
        b.y = Bp[kk * 32 + 32];
        acc = WMMA_F32X4(a, b, acc);
      }
    }
    __syncthreads();
  }
  if (active) {
    // C/D layout: VGPR j -> row m0 + 8*half + j, col n0 + l15
    float* Crow = C + (size_t)(m0 + half * 8) * 32 + n0 + l15;
#pragma unroll
    for (int j = 0; j < 8; ++j) Crow[(size_t)j * 32] = acc[j];
  }
}

// ---------------- group member pooling: gul[g,d] = sum_l mask[g,l]*acc[users[g,l],d] ----
__global__ __launch_bounds__(256) void hrd_members(const int* __restrict__ users,
                                                   const float* __restrict__ mask,
                                                   const float* __restrict__ acc,
                                                   float* __restrict__ gul, int G, int L) {
  int lane = threadIdx.x & 31;
  int g = (blockIdx.x * blockDim.x + threadIdx.x) >> 5;
  if (g >= G) return;
  float s = 0.0f;
  for (int l = 0; l < L; ++l) {
    float w = mask[g * L + l];
    int u = users[g * L + l];
    s += w * acc[(size_t)u * 32 + lane];
  }
  gul[(size_t)g * 32 + lane] = s;
}

// ---------------- gate MLP: w[r] = sigmoid(relu(x @ W1 + b1) @ W2 + b2), wave per row ----
__global__ __launch_bounds__(256) void hrd_gate(const float* __restrict__ gemb,
                                                const float* __restrict__ gul,
                                                const float* __restrict__ w1,
                                                const float* __restrict__ b1,
                                                const float* __restrict__ w2,
                                                const float* __restrict__ b2,
                                                float* __restrict__ wout, int G) {
  int lane = threadIdx.x & 31;
  int r = (blockIdx.x * blockDim.x + threadIdx.x) >> 5;
  if (r >= 2 * G) return;
  const float* xp = (r < G) ? (gemb + (size_t)r * 32) : (gul + (size_t)(r - G) * 32);
  float xv = xp[lane];
  float h = b1[lane];
#pragma unroll
  for (int k = 0; k < 32; ++k) {
    float xk = __shfl(xv, k, 32);
    h += xk * w1[k * 32 + lane];
  }
  h = fmaxf(h, 0.0f);
  float p = h * w2[lane];
#pragma unroll
  for (int off = 16; off > 0; off >>= 1) p += __shfl_xor(p, off, 32);
  if (lane == 0) wout[r] = 1.0f / (1.0f + __expf(-(p + b2[0])));
}

// ---------------- group_final = w[:G]*g_emb + w[G:]*gul ----------------
__global__ __launch_bounds__(256) void hrd_combine(const float* __restrict__ gemb,
                                                   const float* __restrict__ gul,
                                                   const float* __restrict__ w,
                                                   float* __restrict__ gf, int G) {
  int i = blockIdx.x * 256 + threadIdx.x;
  if (i >= G * 32) return;
  int g = i >> 5;
  gf[i] = w[g] * gemb[i] + w[G + g] * gul[i];
}

// ---------------- prediction head: thread per batch element ----------------
__global__ __launch_bounds__(256) void hrd_predict(const int* __restrict__ gidx,
                                                   const int* __restrict__ iidx,
                                                   const float* __restrict__ gf,
                                                   const float* __restrict__ iacc,
                                                   const float* __restrict__ w1,
                                                   const float* __restrict__ b1,
                                                   const float* __restrict__ w2,
                                                   const float* __restrict__ b2,
                                                   float* __restrict__ out, int B) {
  int bi = blockIdx.x * 256 + threadIdx.x;
  if (bi >= B) return;
  const float* ge = gf + (size_t)gidx[bi] * 32;
  const float* ie = iacc + (size_t)iidx[bi] * 32;
  float v[32];
#pragma unroll
  for (int d = 0; d < 32; ++d) v[d] = ge[d] * ie[d];
  float acc = b2[0];
#pragma unroll
  for (int j = 0; j < 8; ++j) {
    float h = b1[j];
#pragma unroll
    for (int d = 0; d < 32; ++d) h += v[d] * w1[d * 8 + j];
    h = fmaxf(h, 0.0f);
    acc += h * w2[j];
  }
  out[bi] = 1.0f / (1.0f + __expf(-acc));
}

// ======================================================================================
extern "C" void kernel_launch(void* const* d_in, const int* in_sizes, int n_in,
                              void* d_out, int out_size, void* d_ws, size_t ws_size,
                              hipStream_t stream) {
  const int* group_inputs   = (const int*)d_in[0];
  const int* item_inputs    = (const int*)d_in[1];
  const float* user_emb     = (const float*)d_in[2];
  const float* group_emb    = (const float*)d_in[3];
  const float* item_emb     = (const float*)d_in[4];
  const int* ui_rows        = (const int*)d_in[5];
  const int* ui_cols        = (const int*)d_in[6];
  const float* ui_vals      = (const float*)d_in[7];
  const int* gi_rows        = (const int*)d_in[8];
  const int* gi_cols        = (const int*)d_in[9];
  const float* gi_vals      = (const float*)d_in[10];
  const float* overlap_user = (const float*)d_in[11];
  const float* overlap_item = (const float*)d_in[12];
  const int* grp_users      = (const int*)d_in[13];
  const float* grp_mask     = (const float*)d_in[14];
  const float* gate_w1      = (const float*)d_in[15];
  const float* gate_b1      = (const float*)d_in[16];
  const float* gate_w2      = (const float*)d_in[17];
  const float* gate_b2      = (const float*)d_in[18];
  const float* pred_w1      = (const float*)d_in[19];
  const float* pred_b1      = (const float*)d_in[20];
  const float* pred_w2      = (const float*)d_in[21];
  const float* pred_b2      = (const float*)d_in[22];

  const int U = 10000, G = 3000, I = 8000, L = 20;
  const int B = in_sizes[0];
  const int NNZ_UI = in_sizes[5];
  const int NNZ_GI = in_sizes[8];
  const int UI32 = (U + I) * 32, GI32 = (G + I) * 32;
  const int U32 = U * 32, I32 = I * 32, G32 = G * 32;

  // bump allocator over workspace
  char* wsbase = (char*)d_ws;
  size_t off = 0;
  auto alloc = [&](size_t nfloat) -> float* {
    float* p = (float*)(wsbase + off);
    off += ((nfloat * sizeof(float) + 255) / 256) * 256;
    return p;
  };
  float* ui0 = alloc(UI32); float* ui1 = alloc(UI32);
  float* ui2 = alloc(UI32); float* ui3 = alloc(UI32);
  float* uefu = alloc(U32); float* iefu = alloc(I32);
  float* gi0 = alloc(GI32); float* gi1 = alloc(GI32);
  float* gi2 = alloc(GI32); float* gi3 = alloc(GI32);
  float* etmp = alloc(GI32);
  float* s1 = alloc(GI32); float* s3 = alloc(GI32);
  float* iemb = alloc(I32); float* gemb = alloc(G32);
  float* tA = alloc(U32);  float* tB = alloc(U32);  float* uacc = alloc(U32);
  float* gul = alloc(G32); float* wgate = alloc(2 * G);
  float* jA = alloc(I32);  float* jB = alloc(I32);  float* iacc = alloc(I32);
  float* gf = alloc(G32);

  auto blocks = [](int n) { return dim3((n + 255) / 256); };
  auto mm = [&](const float* A, const float* X, float* C, int M, int K) {
    int mtiles = (M + 15) / 16;
    hrd_mm_wmma<<<dim3((mtiles + 3) / 4), dim3(256), 0, stream>>>(A, X, C, M, K);
  };
  auto spmm = [&](const int* r, const int* c, const float* v, const float* x,
                  float* o, int on, int nnz) {
    hrd_zero<<<blocks(on), 256, 0, stream>>>(o, on);
    hrd_spmm<<<dim3(1024), dim3(256), 0, stream>>>(r, c, v, x, o, nnz);
  };

  // ---- user-item graph conv (3 layers) ----
  hrd_copy<<<blocks(U32), 256, 0, stream>>>(ui0, user_emb, U32);
  hrd_copy<<<blocks(I32), 256, 0, stream>>>(ui0 + U32, item_emb, I32);
  spmm(ui_rows, ui_cols, ui_vals, ui0, ui1, UI32, NNZ_UI);
  spmm(ui_rows, ui_cols, ui_vals, ui1, ui2, UI32, NNZ_UI);
  spmm(ui_rows, ui_cols, ui_vals, ui2, ui3, UI32, NNZ_UI);
  hrd_mean4<<<blocks(U32), 256, 0, stream>>>(uefu, ui0, ui1, ui2, ui3, U32);
  hrd_mean4<<<blocks(I32), 256, 0, stream>>>(iefu, ui0 + U32, ui1 + U32, ui2 + U32, ui3 + U32, I32);

  // ---- group-item graph conv (3 layers) ----
  hrd_copy<<<blocks(G32), 256, 0, stream>>>(gi0, group_emb, G32);
  hrd_copy<<<blocks(I32), 256, 0, stream>>>(gi0 + G32, iefu, I32);
  spmm(gi_rows, gi_cols, gi_vals, gi0, gi1, GI32, NNZ_GI);
  spmm(gi_rows, gi_cols, gi_vals, gi1, gi2, GI32, NNZ_GI);
  spmm(gi_rows, gi_cols, gi_vals, gi2, gi3, GI32, NNZ_GI);
  hrd_mean4<<<blocks(I32), 256, 0, stream>>>(iemb, gi0 + G32, gi1 + G32, gi2 + G32, gi3 + G32, I32);

  // g_layers odd entries: spmm over concat(group_emb, item_layers[0|2])
  hrd_copy<<<blocks(G32), 256, 0, stream>>>(etmp, group_emb, G32);
  hrd_copy<<<blocks(I32), 256, 0, stream>>>(etmp + G32, ui0 + U32, I32);   // item_layers[0]
  spmm(gi_rows, gi_cols, gi_vals, etmp, s1, GI32, NNZ_GI);
  hrd_copy<<<blocks(G32), 256, 0, stream>>>(etmp, group_emb, G32);
  hrd_copy<<<blocks(I32), 256, 0, stream>>>(etmp + G32, ui2 + U32, I32);   // item_layers[2]
  spmm(gi_rows, gi_cols, gi_vals, etmp, s3, GI32, NNZ_GI);
  hrd_mean4<<<blocks(G32), 256, 0, stream>>>(gemb, gi0, s1, gi2, s3, G32);

  // ---- social power-series on users: acc = ue + O@ue + O@(O@ue)  (WMMA GEMMs) ----
  mm(overlap_user, uefu, tA, U, U);
  mm(overlap_user, tA, tB, U, U);
  hrd_sum3<<<blocks(U32), 256, 0, stream>>>(uacc, uefu, tA, tB, U32);

  // ---- gathered member pooling + gate ----
  hrd_members<<<blocks(G * 32), 256, 0, stream>>>(grp_users, grp_mask, uacc, gul, G, L);
  hrd_gate<<<blocks(2 * G * 32), 256, 0, stream>>>(gemb, gul, gate_w1, gate_b1,
                                                   gate_w2, gate_b2, wgate, G);
  hrd_combine<<<blocks(G32), 256, 0, stream>>>(gemb, gul, wgate, gf, G);

  // ---- social power-series on items (WMMA GEMMs) ----
  mm(overlap_item, iemb, jA, I, I);
  mm(overlap_item, jA, jB, I, I);
  hrd_sum3<<<blocks(I32), 256, 0, stream>>>(iacc, iemb, jA, jB, I32);

  // ---- prediction head ----
  hrd_predict<<<blocks(B), 256, 0, stream>>>(group_inputs, item_inputs, gf, iacc,
                                             pred_w1, pred_b1, pred_w2, pred_b2,
                                             (float*)d_out, B);
}